// GraphRegulator_65481071400876
// MI455X (gfx1250) — compile-verified
//
#include <hip/hip_runtime.h>

typedef __attribute__((ext_vector_type(2))) float v2f;
typedef __attribute__((ext_vector_type(8))) float v8f;

#define BATCH 8
#define NQ    2048
#define DK    64
#define BM    128
#define BN    64
#define LDA   68          // 64 + 4 pad: (4*row + col) % 64 -> all 64 banks distinct
#define THREADS 256

__global__ __launch_bounds__(THREADS) void qgl_zero_deg(float* deg, int n) {
    int i = blockIdx.x * blockDim.x + threadIdx.x;
    if (i < n) deg[i] = 0.0f;
}

__global__ __launch_bounds__(THREADS) void qgl_gemm_lap(const float* __restrict__ S,
                                                        float* __restrict__ out,
                                                        float* __restrict__ deg)
{
    __shared__ float As[BM * LDA];   // 128 rows of states (output rows)
    __shared__ float Bs[BN * LDA];   // 64  rows of states (output cols)

    const int bz = blockIdx.z;       // batch
    const int bm = blockIdx.x;       // 0..15  (2048/128)
    const int bn = blockIdx.y;       // 0..31  (2048/64)

    const float* Sb = S + (size_t)bz * NQ * DK;
    const int tid = threadIdx.x;

    // ---- stage tiles to LDS (float4, coalesced; row*LDA*4 bytes is 16B aligned) ----
    {
        const float* srcA = Sb + (size_t)bm * BM * DK;
        #pragma unroll
        for (int i = 0; i < 8; ++i) {                 // 128*16 float4 / 256 thr
            int idx = tid + i * THREADS;
            int row = idx >> 4, c4 = idx & 15;
            *(float4*)&As[row * LDA + c4 * 4] = *(const float4*)(srcA + row * DK + c4 * 4);
        }
        const float* srcB = Sb + (size_t)bn * BN * DK;
        #pragma unroll
        for (int i = 0; i < 4; ++i) {                 // 64*16 float4 / 256 thr
            int idx = tid + i * THREADS;
            int row = idx >> 4, c4 = idx & 15;
            *(float4*)&Bs[row * LDA + c4 * 4] = *(const float4*)(srcB + row * DK + c4 * 4);
        }
    }
    __syncthreads();

    const int lane = tid & 31;
    const int wave = tid >> 5;        // 0..7
    const int half = lane >> 4;       // 0/1
    const int l16  = lane & 15;
    const int wm   = wave >> 1;       // 0..3 -> M offset wm*32
    const int wn   = wave & 1;        // 0..1 -> N offset wn*32

    // ---- K=64 fully resident: 16 chained V_WMMA_F32_16X16X4_F32 per tile ----
    v8f acc[2][2] = {};
    const int aBase0 = (wm * 32 + l16) * LDA;
    const int aBase1 = (wm * 32 + 16 + l16) * LDA;
    const int bBase0 = (wn * 32 + l16) * LDA;
    const int bBase1 = (wn * 32 + 16 + l16) * LDA;
    #pragma unroll
    for (int k = 0; k < 16; ++k) {
        const int co = 4 * k + 2 * half;  // ISA 16x4 fp32 layout: low lanes K={0,1}, high lanes K={2,3}
        v2f a0 = *(const v2f*)&As[aBase0 + co];
        v2f a1 = *(const v2f*)&As[aBase1 + co];
        v2f b0 = *(const v2f*)&Bs[bBase0 + co];
        v2f b1 = *(const v2f*)&Bs[bBase1 + co];
        acc[0][0] = __builtin_amdgcn_wmma_f32_16x16x4_f32(false, a0, false, b0, (short)0, acc[0][0], false, false);
        acc[0][1] = __builtin_amdgcn_wmma_f32_16x16x4_f32(false, a0, false, b1, (short)0, acc[0][1], false, false);
        acc[1][0] = __builtin_amdgcn_wmma_f32_16x16x4_f32(false, a1, false, b0, (short)0, acc[1][0], false, false);
        acc[1][1] = __builtin_amdgcn_wmma_f32_16x16x4_f32(false, a1, false, b1, (short)0, acc[1][1], false, false);
    }

    // ---- fuse: fid = d*d, threshold, zero diag, write -w, accumulate degree ----
    float* outB = out + (size_t)bz * NQ * NQ;
    #pragma unroll
    for (int mt = 0; mt < 2; ++mt) {
        const int growBase = bm * BM + wm * 32 + mt * 16 + 8 * half;  // + v
        float rsum[8];
        #pragma unroll
        for (int v = 0; v < 8; ++v) rsum[v] = 0.0f;

        #pragma unroll
        for (int nt = 0; nt < 2; ++nt) {
            const int gcol = bn * BN + wn * 32 + nt * 16 + l16;
            #pragma unroll
            for (int v = 0; v < 8; ++v) {
                const int grow = growBase + v;
                const float d  = acc[mt][nt][v];
                const float f  = d * d;
                float w = (f >= 0.95f) ? 1.0f : ((f >= 0.5f) ? 0.5f : 0.0f);
                if (grow == gcol) w = 0.0f;
                outB[(size_t)grow * NQ + gcol] = (w > 0.0f) ? -w : 0.0f;
                rsum[v] += w;
            }
        }
        // butterfly reduce across the 16-lane half (masks 1..8 stay in-half)
        #pragma unroll
        for (int m = 1; m <= 8; m <<= 1) {
            #pragma unroll
            for (int v = 0; v < 8; ++v)
                rsum[v] += __shfl_xor(rsum[v], m, 32);
        }
        if (l16 == 0) {
            #pragma unroll
            for (int v = 0; v < 8; ++v)
                atomicAdd(&deg[bz * NQ + growBase + v], rsum[v]);
        }
    }
}

__global__ __launch_bounds__(THREADS) void qgl_write_diag(float* __restrict__ out,
                                                          const float* __restrict__ deg)
{
    int i = blockIdx.x * blockDim.x + threadIdx.x;   // 0 .. BATCH*NQ-1
    if (i < BATCH * NQ) {
        int b = i / NQ, r = i - b * NQ;
        out[(size_t)b * NQ * NQ + (size_t)r * NQ + r] = deg[i];
    }
}

extern "C" void kernel_launch(void* const* d_in, const int* in_sizes, int n_in,
                              void* d_out, int out_size, void* d_ws, size_t ws_size,
                              hipStream_t stream) {
    (void)in_sizes; (void)n_in; (void)out_size; (void)ws_size;
    const float* S = (const float*)d_in[0];
    float* out = (float*)d_out;
    float* deg = (float*)d_ws;                 // BATCH*NQ floats = 64 KiB scratch

    const int nd = BATCH * NQ;
    qgl_zero_deg<<<(nd + THREADS - 1) / THREADS, THREADS, 0, stream>>>(deg, nd);

    dim3 grid(NQ / BM, NQ / BN, BATCH);        // (16, 32, 8)
    qgl_gemm_lap<<<grid, THREADS, 0, stream>>>(S, out, deg);

    qgl_write_diag<<<(nd + THREADS - 1) / THREADS, THREADS, 0, stream>>>(out, deg);
}